// GCNFold_9010841387502
// MI455X (gfx1250) — compile-verified
//
#include <hip/hip_runtime.h>
#include <math.h>

#define BDIM 2
#define LSEQ 512
#define EDIM 640
#define HDIM 256
#define NLAY 3

#define KC   32            // K-chunk staged per LDS round
#define NPAIR (KC / 2)     // row pairs per chunk = 16
#define PSTR 160           // dwords per LDS row-pair: 64*2 data + 32 pad (160%64==32)

typedef __attribute__((ext_vector_type(2))) float v2f;
typedef __attribute__((ext_vector_type(8))) float v8f;

// ---------------------------------------------------------------------------
// Build adjacency A[b,s,j] (fp32 0/1) from bp_probs, locality, lengths.
// ---------------------------------------------------------------------------
__global__ void build_adj_kernel(const float* __restrict__ bp,
                                 const int* __restrict__ lengths,
                                 float* __restrict__ A) {
  long idx = (long)blockIdx.x * blockDim.x + threadIdx.x;
  const long total = (long)BDIM * LSEQ * LSEQ;
  if (idx >= total) return;
  int j = (int)(idx % LSEQ);
  int s = (int)((idx / LSEQ) % LSEQ);
  int b = (int)(idx / ((long)LSEQ * LSEQ));
  int diff = s > j ? s - j : j - s;
  bool local = (diff >= 1) && (diff <= 2);
  bool dist  = diff >= 3;
  bool edge  = local || (dist && (bp[idx] > 0.01f));
  int len = lengths[b];
  edge = edge && (s < len) && (j < len);
  A[idx] = edge ? 1.0f : 0.0f;
}

// ---------------------------------------------------------------------------
// Row (out) and column (in) degree norms: rsqrt(max(sum,1)).
// One 128-thread block per (b, index r); computes both row r and column r.
// ---------------------------------------------------------------------------
__global__ void __launch_bounds__(128)
degree_kernel(const float* __restrict__ A,
              float* __restrict__ out_norm,
              float* __restrict__ in_norm) {
  __shared__ float red[128];
  int b = blockIdx.y;
  int r = blockIdx.x;
  const float* Ab = A + (long)b * LSEQ * LSEQ;
  float rowsum = 0.f, colsum = 0.f;
  for (int t = threadIdx.x; t < LSEQ; t += 128) {
    rowsum += Ab[(long)r * LSEQ + t];
    colsum += Ab[(long)t * LSEQ + r];
  }
  red[threadIdx.x] = rowsum;
  __syncthreads();
  for (int off = 64; off > 0; off >>= 1) {
    if (threadIdx.x < off) red[threadIdx.x] += red[threadIdx.x + off];
    __syncthreads();
  }
  float rs = red[0];
  __syncthreads();
  red[threadIdx.x] = colsum;
  __syncthreads();
  for (int off = 64; off > 0; off >>= 1) {
    if (threadIdx.x < off) red[threadIdx.x] += red[threadIdx.x + off];
    __syncthreads();
  }
  if (threadIdx.x == 0) {
    out_norm[(long)b * LSEQ + r] = rsqrtf(fmaxf(rs, 1.0f));
    in_norm [(long)b * LSEQ + r] = rsqrtf(fmaxf(red[0], 1.0f));
  }
}

// ---------------------------------------------------------------------------
// FP32 WMMA GEMM: C = act(opA(A) @ B + bias).
// Block = 128 threads = 4 waves. Block tile = 64M x 64N:
//   wave w owns rows [m0 + 16w, +16), all 64 N columns (4 accumulators).
// B panel (KC x 64) shared across waves, staged in LDS *interleaved by row
// pair*:  sB[pair*PSTR + n*2 + v] = B[k0 + 2*pair + v][n0 + n]
// so each WMMA B-operand (rows 2p,2p+1 at one column) is ONE aligned
// ds_load_b64 straight into a register pair (no v_mov fixups).
// A operands for the whole chunk are prefetched into av[8] (one load clause)
// before the wmma drain loop, hiding global latency behind LDS reads.
// V_WMMA_F32_16X16X4_F32 operand layouts (wave32):
//   A 16x4 : lane l holds (m = l&15, k = 2*(l>>4) + v), v in {0,1}
//   B 4x16 : lane l holds (k = 2*(l>>4) + v, n = l&15)
//   C 16x16: vgpr r, lane l -> (m = r + 8*(l>>4), n = l&15)
// TRANSA: A stored K-major (element (m,k) = A[k*lda + m]) -> computes A^T @ B.
// Requires: M % 64 == 0, N % 64 == 0, K % KC == 0 (true for all call sites).
// ---------------------------------------------------------------------------
template <bool TRANSA, bool RELU>
__global__ void __launch_bounds__(128)
wmma_gemm_kernel(const float* __restrict__ A, int lda, long sA,
                 const float* __restrict__ B, int ldb, long sB_,
                 const float* __restrict__ bias,
                 float* __restrict__ C, int ldc, long sC, int K) {
  __shared__ float sB[NPAIR * PSTR];  // 16*160*4 = 10240 B
  int tid  = threadIdx.x;
  int lane = tid & 31;
  int wave = tid >> 5;
  long z = blockIdx.z;
  A += sA * z; B += sB_ * z; C += sC * z;

  int n0 = blockIdx.x * 64;
  int m0 = blockIdx.y * 64 + wave * 16;

  int ml = lane & 15;
  int kh = (lane >> 4) << 1;  // 0 or 2

  v8f acc0 = {0.f,0.f,0.f,0.f,0.f,0.f,0.f,0.f};
  v8f acc1 = acc0, acc2 = acc0, acc3 = acc0;

  for (int k0 = 0; k0 < K; k0 += KC) {
    // ---- stage B[k0:k0+KC, n0:n0+64] into LDS, interleaved by row pair ----
    // 256 units: unit u -> pair = u>>4, 4-col group c = (u&15)*4.
    for (int u = tid; u < NPAIR * 16; u += 128) {
      int pair = u >> 4;
      int c    = (u & 15) << 2;
      const float* src0 = B + (long)(k0 + 2 * pair) * ldb + n0 + c;
      const float4 r0 = *(const float4*)src0;
      const float4 r1 = *(const float4*)(src0 + ldb);
      float* dst = &sB[pair * PSTR + c * 2];
      float4 lo = {r0.x, r1.x, r0.y, r1.y};
      float4 hi = {r0.z, r1.z, r0.w, r1.w};
      *(float4*)(dst + 0) = lo;
      *(float4*)(dst + 4) = hi;
    }
    __syncthreads();

    // ---- prefetch all A operands for this chunk (one load clause) ----
    v2f av[KC / 4];
#pragma unroll
    for (int s = 0; s < KC / 4; ++s) {
      int ka = k0 + 4 * s + kh;
      if (TRANSA) {
        const float* ap = A + (long)ka * lda + (m0 + ml);
        av[s].x = ap[0];
        av[s].y = ap[lda];
      } else {
        av[s] = *(const v2f*)(A + (long)(m0 + ml) * lda + ka);
      }
    }

    // ---- drain: 4 wmmas per K-step, B operands as single ds_load_b64 ----
#pragma unroll
    for (int s = 0; s < KC / 4; ++s) {
      int pair = (4 * s + kh) >> 1;  // rows (4s+kh, 4s+kh+1)
      const float* base = &sB[pair * PSTR + ml * 2];
      v2f b0 = *(const v2f*)(base + 0);
      v2f b1 = *(const v2f*)(base + 32);
      v2f b2 = *(const v2f*)(base + 64);
      v2f b3 = *(const v2f*)(base + 96);
      acc0 = __builtin_amdgcn_wmma_f32_16x16x4_f32(false, av[s], false, b0, (short)0, acc0, false, false);
      acc1 = __builtin_amdgcn_wmma_f32_16x16x4_f32(false, av[s], false, b1, (short)0, acc1, false, false);
      acc2 = __builtin_amdgcn_wmma_f32_16x16x4_f32(false, av[s], false, b2, (short)0, acc2, false, false);
      acc3 = __builtin_amdgcn_wmma_f32_16x16x4_f32(false, av[s], false, b3, (short)0, acc3, false, false);
    }
    __syncthreads();
  }

  // ---- epilogue: bias + optional relu, 4 N-tiles ----
  int mbase = m0 + ((lane >> 4) << 3);
  v8f accs[4] = {acc0, acc1, acc2, acc3};
#pragma unroll
  for (int t = 0; t < 4; ++t) {
    int n = n0 + 16 * t + ml;
    float bval = bias ? bias[n] : 0.0f;
#pragma unroll
    for (int r = 0; r < 8; ++r) {
      float v = accs[t][r] + bval;
      if (RELU) v = fmaxf(v, 0.0f);
      C[(long)(mbase + r) * ldc + n] = v;
    }
  }
}

// ---------------------------------------------------------------------------
// Y[i] = X[i] * scale[i / HDIM]   (per-row scaling of (B*L, H) features)
// ---------------------------------------------------------------------------
__global__ void scale_rows_kernel(const float* __restrict__ X,
                                  const float* __restrict__ s,
                                  float* __restrict__ Y, long total) {
  long i = (long)blockIdx.x * blockDim.x + threadIdx.x;
  if (i < total) Y[i] = X[i] * s[i / HDIM];
}

// ---------------------------------------------------------------------------
// Edge predictor, fused over 16x16 (i,j) tiles:
//   logit = relu(pi[i,:] + pj[j,:] + b1) . w2 + b2, masked BEFORE sigmoid,
//   out = (1-sigmoid(prior)) * sigmoid(logit) + sigmoid(prior) * bp
// pi/pj tiles + b1/w2 staged in LDS (34 KB).
// ---------------------------------------------------------------------------
__global__ void __launch_bounds__(256)
edge_kernel(const float* __restrict__ pi, const float* __restrict__ pj,
            const float* __restrict__ ep_b1, const float* __restrict__ ep_w2,
            const float* __restrict__ ep_b2, const float* __restrict__ prior,
            const float* __restrict__ bp, const int* __restrict__ lengths,
            float* __restrict__ out) {
  __shared__ float spi[16 * HDIM];
  __shared__ float spj[16 * HDIM];
  __shared__ float sb1[HDIM];
  __shared__ float sw2[HDIM];
  int b = blockIdx.z;
  int i0 = blockIdx.y * 16, j0 = blockIdx.x * 16;
  int tid = threadIdx.x;
  const float* pib = pi + ((long)b * LSEQ + i0) * HDIM;
  const float* pjb = pj + ((long)b * LSEQ + j0) * HDIM;
  for (int t = tid; t < 16 * HDIM; t += 256) {
    spi[t] = pib[t];
    spj[t] = pjb[t];
  }
  for (int t = tid; t < HDIM; t += 256) {
    sb1[t] = ep_b1[t];
    sw2[t] = ep_w2[t];
  }
  __syncthreads();

  int il = tid >> 4, jl = tid & 15;
  const float* pr = &spi[il * HDIM];
  const float* qr = &spj[jl * HDIM];
  float acc = 0.f;
#pragma unroll 4
  for (int hh = 0; hh < HDIM; ++hh) {
    float zv = pr[hh] + qr[hh] + sb1[hh];
    acc = fmaf(fmaxf(zv, 0.f), sw2[hh], acc);
  }
  float logit = acc + ep_b2[0];

  int i = i0 + il, j = j0 + jl;
  int diff = i > j ? i - j : j - i;
  int len = lengths[b];
  bool mask = (diff >= 3) && (i < len) && (j < len);
  if (!mask) logit = 0.0f;  // logits *= mask, before sigmoid

  float w = 1.0f / (1.0f + expf(-prior[0]));
  float sg = 1.0f / (1.0f + expf(-logit));
  long idx = ((long)b * LSEQ + i) * LSEQ + j;
  out[idx] = (1.0f - w) * sg + w * bp[idx];
}

// ---------------------------------------------------------------------------
extern "C" void kernel_launch(void* const* d_in, const int* in_sizes, int n_in,
                              void* d_out, int out_size, void* d_ws, size_t ws_size,
                              hipStream_t stream) {
  const float* emb     = (const float*)d_in[0];   // (B,L,E)
  const float* bp      = (const float*)d_in[1];   // (B,L,L)
  const int*   lengths = (const int*)  d_in[2];   // (B,)
  const float* mlp_w1  = (const float*)d_in[3];   // (E, 2H)
  const float* mlp_b1  = (const float*)d_in[4];   // (2H,)
  const float* mlp_w2  = (const float*)d_in[5];   // (2H, H)
  const float* mlp_b2  = (const float*)d_in[6];   // (H,)
  const float* gcn_w   = (const float*)d_in[7];   // (NLAY, H, H)
  const float* gcn_b   = (const float*)d_in[8];   // (NLAY, H)
  const float* ep_w1   = (const float*)d_in[9];   // (2H, H)
  const float* ep_b1   = (const float*)d_in[10];  // (H,)
  const float* ep_w2   = (const float*)d_in[11];  // (H, 1)
  const float* ep_b2   = (const float*)d_in[12];  // (1,)
  const float* prior   = (const float*)d_in[13];  // scalar

  float* out = (float*)d_out;

  // Workspace layout (floats)
  float* ws   = (float*)d_ws;
  float* Adj  = ws;                                   // B*L*L      = 524288
  float* onrm = Adj  + (long)BDIM * LSEQ * LSEQ;      // B*L        = 1024
  float* inrm = onrm + (long)BDIM * LSEQ;             // B*L        = 1024
  float* X1   = inrm + (long)BDIM * LSEQ;             // B*L*2H     = 524288
  float* h    = X1   + (long)BDIM * LSEQ * 2 * HDIM;  // B*L*H      = 262144
  float* hs   = h    + (long)BDIM * LSEQ * HDIM;      // B*L*H
  float* agg  = hs   + (long)BDIM * LSEQ * HDIM;      // B*L*H
  float* pi   = agg  + (long)BDIM * LSEQ * HDIM;      // B*L*H
  float* pj   = pi   + (long)BDIM * LSEQ * HDIM;      // B*L*H

  const int  ML   = BDIM * LSEQ;  // 1024 rows flattened
  const long totA = (long)BDIM * LSEQ * LSEQ;
  const long totH = (long)ML * HDIM;

  // 1) adjacency + degree norms
  build_adj_kernel<<<dim3((unsigned)((totA + 255) / 256)), 256, 0, stream>>>(bp, lengths, Adj);
  degree_kernel<<<dim3(LSEQ, BDIM), 128, 0, stream>>>(Adj, onrm, inrm);

  // 2) MLP: X1 = relu(emb @ w1 + b1)   (1024x640)@(640x512)
  wmma_gemm_kernel<false, true><<<dim3((2 * HDIM) / 64, ML / 64, 1), 128, 0, stream>>>(
      emb, EDIM, 0, mlp_w1, 2 * HDIM, 0, mlp_b1, X1, 2 * HDIM, 0, EDIM);
  //    h = X1 @ w2 + b2               (1024x512)@(512x256), no relu
  wmma_gemm_kernel<false, false><<<dim3(HDIM / 64, ML / 64, 1), 128, 0, stream>>>(
      X1, 2 * HDIM, 0, mlp_w2, HDIM, 0, mlp_b2, h, HDIM, 0, 2 * HDIM);

  // 3) GCN layers
  for (int l = 0; l < NLAY; ++l) {
    scale_rows_kernel<<<dim3((unsigned)((totH + 255) / 256)), 256, 0, stream>>>(h, onrm, hs, totH);
    // agg[b] = Adj[b]^T @ hs[b]   (512x512)^T @ (512x256), batched over B
    wmma_gemm_kernel<true, false><<<dim3(HDIM / 64, LSEQ / 64, BDIM), 128, 0, stream>>>(
        Adj, LSEQ, (long)LSEQ * LSEQ, hs, HDIM, (long)LSEQ * HDIM, nullptr,
        agg, HDIM, (long)LSEQ * HDIM, LSEQ);
    scale_rows_kernel<<<dim3((unsigned)((totH + 255) / 256)), 256, 0, stream>>>(agg, inrm, agg, totH);
    // h = relu(agg @ gcn_w[l] + gcn_b[l])   (1024x256)@(256x256)
    wmma_gemm_kernel<false, true><<<dim3(HDIM / 64, ML / 64, 1), 128, 0, stream>>>(
        agg, HDIM, 0, gcn_w + (long)l * HDIM * HDIM, HDIM, 0, gcn_b + (long)l * HDIM,
        h, HDIM, 0, HDIM);
  }

  // 4) projections: pi = h @ ep_w1[:H], pj = h @ ep_w1[H:]
  wmma_gemm_kernel<false, false><<<dim3(HDIM / 64, ML / 64, 1), 128, 0, stream>>>(
      h, HDIM, 0, ep_w1, HDIM, 0, nullptr, pi, HDIM, 0, HDIM);
  wmma_gemm_kernel<false, false><<<dim3(HDIM / 64, ML / 64, 1), 128, 0, stream>>>(
      h, HDIM, 0, ep_w1 + (long)HDIM * HDIM, HDIM, 0, nullptr, pj, HDIM, 0, HDIM);

  // 5) fused edge predictor + prior mix
  edge_kernel<<<dim3(LSEQ / 16, LSEQ / 16, BDIM), 256, 0, stream>>>(
      pi, pj, ep_b1, ep_w2, ep_b2, prior, bp, lengths, out);
}